// AttentionModel_26079041422171
// MI455X (gfx1250) — compile-verified
//
#include <hip/hip_runtime.h>
#include <math.h>

// Problem constants (match reference)
#define B_     1024
#define N_     200
#define D_     256
#define H_     8
#define START_ 24
#define NEG_   (-1000000000.0f)
#define BT     16   // batches per step-kernel block

typedef __attribute__((ext_vector_type(2))) float v2f;
typedef __attribute__((ext_vector_type(8))) float v8f;

// D = A(16x4) * B(4x16) + C, fp32 WMMA (CDNA5)
__device__ __forceinline__ v8f wmma_f32(v2f a, v2f b, v8f c) {
  return __builtin_amdgcn_wmma_f32_16x16x4_f32(false, a, false, b, (short)0, c,
                                               false, false);
}

// ---------------------------------------------------------------------------
// State init: visited[b][n] = (n==START), prev[b] = START
// ---------------------------------------------------------------------------
__global__ void init_state_kernel(unsigned char* __restrict__ visited,
                                  int* __restrict__ prev) {
  size_t i = (size_t)blockIdx.x * blockDim.x + threadIdx.x;
  if (i < (size_t)B_ * N_) visited[i] = ((i % N_) == START_) ? 1 : 0;
  if (i < B_) prev[i] = START_;
}

// ---------------------------------------------------------------------------
// meanE[b][d] = mean_n(embeddings[b][n][d]) + pref[d]
// ---------------------------------------------------------------------------
__global__ void mean_kernel(const float* __restrict__ E,
                            const float* __restrict__ pref,
                            float* __restrict__ meanE) {
  int b = blockIdx.x, dd = threadIdx.x;
  const float* p = E + (size_t)b * N_ * D_ + dd;
  float s = 0.f;
  for (int n = 0; n < N_; ++n) s += p[(size_t)n * D_];
  meanE[(size_t)b * D_ + dd] = s * (1.0f / N_) + pref[dd];
}

// ---------------------------------------------------------------------------
// fp32 WMMA GEMM: Dst[M x Ncols] = (A + abias_broadcast) @ Bm[256 x Ncols]
// K is fixed at 256. Block = 256 threads (8 waves); block covers a 16-row x
// 128-col output slab. A tile (16x256) is staged once in LDS with the bias
// add fused, so the WMMA inner loop is branch-free (ds_load A, global B).
// grid = (M/16, Ncols/128).
// ---------------------------------------------------------------------------
__global__ __launch_bounds__(256) void wmma_gemm_kernel(
    const float* __restrict__ A, const float* __restrict__ Bm,
    const float* __restrict__ abias, float* __restrict__ Dst, int Ncols) {
  __shared__ __align__(16) float As[16 * 256];
  const int K = 256;
  int tid = threadIdx.x;
  size_t rowBase = (size_t)blockIdx.x * 16;

  // stage A tile (+ broadcast bias over k), 16 floats per thread
  if (abias) {
    for (int idx = tid; idx < 16 * 256; idx += 256) {
      int r = idx >> 8, k = idx & 255;
      As[idx] = A[(rowBase + r) * (size_t)K + k] + abias[k];
    }
  } else {
    for (int idx = tid; idx < 16 * 256; idx += 256) {
      int r = idx >> 8, k = idx & 255;
      As[idx] = A[(rowBase + r) * (size_t)K + k];
    }
  }
  __syncthreads();

  int w = tid >> 5, lane = tid & 31;
  int hi = lane >> 4, l15 = lane & 15;
  int colBase = blockIdx.y * 128 + w * 16;
  v8f c = {};
  for (int k = 0; k < K; k += 4) {
    int ka = k + 2 * hi;
    v2f a, b;
    a.x = As[l15 * 256 + ka];
    a.y = As[l15 * 256 + ka + 1];
    b.x = Bm[(size_t)ka * Ncols + colBase + l15];
    b.y = Bm[(size_t)(ka + 1) * Ncols + colBase + l15];
    c = wmma_f32(a, b, c);
  }
#pragma unroll
  for (int i = 0; i < 8; ++i)
    Dst[(rowBase + i + 8 * hi) * (size_t)Ncols + colBase + l15] = c[i];
}

// ---------------------------------------------------------------------------
// One decode step. 64 blocks x 256 threads (8 waves); block owns 16 batches.
// gKVL layout: row (b*N+n) of 768 floats = [gK(256) | gV(256) | lK(256)].
// ---------------------------------------------------------------------------
__global__ __launch_bounds__(256) void step_kernel(
    const float* __restrict__ E, const float* __restrict__ pref,
    const float* __restrict__ Wstep, const float* __restrict__ Wout,
    const float* __restrict__ gKVL, const float* __restrict__ fixedc,
    int* __restrict__ prev, unsigned char* __restrict__ visited,
    float* __restrict__ out, int t) {
  // 64KB LDS, phase-disjoint aliasing:
  //  [0,16K)  qs   : q (16x256)            (phase1 write, phase2 read)
  //  [16,32K) glim : glimpse (16x256)      (phase2 write, phase3 read)
  //           prevS aliases glim start     (staging only)
  //  [32,64K) regR : X (16x512) phase0/1 | attn rows phase2 | gout phase3/4
  __shared__ __align__(16) char smem[65536];
  float* qs = (float*)smem;
  float* glim = (float*)(smem + 16384);
  char* regR = smem + 32768;
  float* Xs = (float*)regR;
  float* attnR = (float*)regR;
  float* gout = (float*)regR;
  int* prevS = (int*)glim;

  int tid = threadIdx.x;
  int w = tid >> 5;
  int lane = tid & 31;
  int hi = lane >> 4, l15 = lane & 15;
  int bBase = blockIdx.x * BT;

  if (tid < BT) prevS[tid] = prev[bBase + tid];
  __syncthreads();

  // ---- phase 0: stage X = [first_emb | cur_emb] (16 x 512) ----
  for (int idx = tid; idx < BT * 2 * D_; idx += 256) {
    int bl = idx >> 9;
    int k = idx & 511;
    int b = bBase + bl;
    float v;
    if (k < D_)
      v = E[((size_t)b * N_ + START_) * D_ + k] + pref[k];
    else {
      int kk = k - D_;
      v = E[((size_t)b * N_ + prevS[bl]) * D_ + kk] + pref[kk];
    }
    Xs[idx] = v;
  }
  __syncthreads();

  // ---- phase 1: q = fixed_ctx + X @ Wstep  (16x512 @ 512x256), WMMA ----
  {
    int ct0 = w * 2, ct1 = w * 2 + 1;
    v8f c0, c1;
#pragma unroll
    for (int i = 0; i < 8; ++i) {
      size_t r = (size_t)(bBase + i + 8 * hi) * D_;
      c0[i] = fixedc[r + ct0 * 16 + l15];
      c1[i] = fixedc[r + ct1 * 16 + l15];
    }
    for (int k = 0; k < 2 * D_; k += 4) {
      int ka = k + 2 * hi;
      v2f a, b0, b1;
      a.x = Xs[l15 * 512 + ka];
      a.y = Xs[l15 * 512 + ka + 1];
      b0.x = Wstep[(size_t)ka * D_ + ct0 * 16 + l15];
      b0.y = Wstep[(size_t)(ka + 1) * D_ + ct0 * 16 + l15];
      b1.x = Wstep[(size_t)ka * D_ + ct1 * 16 + l15];
      b1.y = Wstep[(size_t)(ka + 1) * D_ + ct1 * 16 + l15];
      c0 = wmma_f32(a, b0, c0);
      c1 = wmma_f32(a, b1, c1);
    }
#pragma unroll
    for (int i = 0; i < 8; ++i) {
      qs[(i + 8 * hi) * D_ + ct0 * 16 + l15] = c0[i];
      qs[(i + 8 * hi) * D_ + ct1 * 16 + l15] = c1[i];
    }
  }
  __syncthreads();

  // ---- phase 2: masked multi-head attention + glimpse ----
  const float inv_sqrt_d = 0.17677669529663687f;  // 1/sqrt(32)
  for (int pb = 0; pb < 2; ++pb) {
    int bl = w * 2 + pb;
    int b = bBase + bl;
    for (int h = 0; h < H_; ++h) {
      float qv[32];
#pragma unroll
      for (int dd = 0; dd < 32; ++dd) qv[dd] = qs[bl * D_ + h * 32 + dd];
      float cj[7];
      float m = -3.0e38f;
#pragma unroll
      for (int j = 0; j < 7; ++j) {
        int n = lane + 32 * j;
        float cv = -3.0e38f;
        if (n < N_) {
          const float4* kp =
              (const float4*)(gKVL + ((size_t)b * N_ + n) * 768 + h * 32);
          float dot = 0.f;
#pragma unroll
          for (int q4 = 0; q4 < 8; ++q4) {
            float4 kv = kp[q4];
            dot += kv.x * qv[4 * q4] + kv.y * qv[4 * q4 + 1] +
                   kv.z * qv[4 * q4 + 2] + kv.w * qv[4 * q4 + 3];
          }
          cv = dot * inv_sqrt_d;
          if (visited[(size_t)b * N_ + n]) cv = NEG_;
          m = fmaxf(m, cv);
        }
        cj[j] = cv;
      }
      for (int off = 16; off > 0; off >>= 1)
        m = fmaxf(m, __shfl_xor(m, off, 32));
      float s = 0.f;
#pragma unroll
      for (int j = 0; j < 7; ++j) {
        int n = lane + 32 * j;
        if (n < N_) {
          float e = expf(cj[j] - m);
          cj[j] = e;
          s += e;
        }
      }
      for (int off = 16; off > 0; off >>= 1) s += __shfl_xor(s, off, 32);
      float inv = 1.0f / s;
      float* arow = attnR + w * 208;  // per-wave private row
#pragma unroll
      for (int j = 0; j < 7; ++j) {
        int n = lane + 32 * j;
        if (n < N_) arow[n] = cj[j] * inv;
      }
      __syncthreads();  // uniform across all 8 waves
      // glimpse: lane = head-dim
      float g = 0.f;
      const float* vp = gKVL + ((size_t)b * N_) * 768 + D_ + h * 32 + lane;
      for (int n = 0; n < N_; ++n) g += arow[n] * vp[(size_t)n * 768];
      glim[bl * D_ + h * 32 + lane] = g;
      __syncthreads();
    }
  }
  __syncthreads();

  // ---- phase 3: gout = glim @ Wout (16x256 @ 256x256), WMMA ----
  {
    int ct0 = w * 2, ct1 = w * 2 + 1;
    v8f c0 = {}, c1 = {};
    for (int k = 0; k < D_; k += 4) {
      int ka = k + 2 * hi;
      v2f a, b0, b1;
      a.x = glim[l15 * D_ + ka];
      a.y = glim[l15 * D_ + ka + 1];
      b0.x = Wout[(size_t)ka * D_ + ct0 * 16 + l15];
      b0.y = Wout[(size_t)(ka + 1) * D_ + ct0 * 16 + l15];
      b1.x = Wout[(size_t)ka * D_ + ct1 * 16 + l15];
      b1.y = Wout[(size_t)(ka + 1) * D_ + ct1 * 16 + l15];
      c0 = wmma_f32(a, b0, c0);
      c1 = wmma_f32(a, b1, c1);
    }
#pragma unroll
    for (int i = 0; i < 8; ++i) {
      gout[(i + 8 * hi) * D_ + ct0 * 16 + l15] = c0[i];
      gout[(i + 8 * hi) * D_ + ct1 * 16 + l15] = c1[i];
    }
  }
  __syncthreads();

  // ---- phase 4: logits, tanh clip, masked log_softmax, argmax, update ----
  const float inv_sqrt_D = 0.0625f;  // 1/sqrt(256)
  for (int pb = 0; pb < 2; ++pb) {
    int bl = w * 2 + pb;
    int b = bBase + bl;
    float lj[7];
    float m = -3.0e38f;
#pragma unroll
    for (int j = 0; j < 7; ++j) {
      int n = lane + 32 * j;
      float lv = -3.0e38f;
      if (n < N_) {
        const float4* kp =
            (const float4*)(gKVL + ((size_t)b * N_ + n) * 768 + 2 * D_);
        float dot = 0.f;
        for (int q4 = 0; q4 < 64; ++q4) {
          float4 kv = kp[q4];
          dot += kv.x * gout[bl * D_ + 4 * q4] +
                 kv.y * gout[bl * D_ + 4 * q4 + 1] +
                 kv.z * gout[bl * D_ + 4 * q4 + 2] +
                 kv.w * gout[bl * D_ + 4 * q4 + 3];
        }
        lv = 10.0f * tanhf(dot * inv_sqrt_D);
        if (visited[(size_t)b * N_ + n]) lv = NEG_;
        m = fmaxf(m, lv);
      }
      lj[j] = lv;
    }
    for (int off = 16; off > 0; off >>= 1)
      m = fmaxf(m, __shfl_xor(m, off, 32));
    float s = 0.f;
#pragma unroll
    for (int j = 0; j < 7; ++j) {
      int n = lane + 32 * j;
      if (n < N_) s += expf(lj[j] - m);
    }
    for (int off = 16; off > 0; off >>= 1) s += __shfl_xor(s, off, 32);
    float lse = logf(s);
    float bv = -3.0e38f;
    int bi = 0x7fffffff;
#pragma unroll
    for (int j = 0; j < 7; ++j) {
      int n = lane + 32 * j;
      if (n < N_) {
        out[((size_t)b * (N_ - 1) + t) * N_ + n] = lj[j] - m - lse;
        if (lj[j] > bv) { bv = lj[j]; bi = n; }  // first-index tie within lane
      }
    }
    for (int off = 16; off > 0; off >>= 1) {
      float ov = __shfl_xor(bv, off, 32);
      int oi = __shfl_xor(bi, off, 32);
      if (ov > bv || (ov == bv && oi < bi)) { bv = ov; bi = oi; }
    }
    if (lane == 0) {
      visited[(size_t)b * N_ + bi] = 1;
      prev[b] = bi;
    }
  }
}

// ---------------------------------------------------------------------------
extern "C" void kernel_launch(void* const* d_in, const int* in_sizes, int n_in,
                              void* d_out, int out_size, void* d_ws,
                              size_t ws_size, hipStream_t stream) {
  const float* E = (const float*)d_in[0];      // embeddings (B,N,D)
  const float* pref = (const float*)d_in[1];   // pref_embed (D)
  const float* Wnode = (const float*)d_in[2];  // (D, 3D)
  const float* Wfixed = (const float*)d_in[3]; // (D, D)
  const float* Wstep = (const float*)d_in[4];  // (2D, D)
  const float* Wout = (const float*)d_in[5];   // (D, D)
  float* out = (float*)d_out;
  char* ws = (char*)d_ws;
  (void)in_sizes; (void)n_in; (void)out_size; (void)ws_size;

  size_t off = 0;
  float* gKVL = (float*)(ws + off);  off += (size_t)B_ * N_ * 768 * sizeof(float);
  float* meanE = (float*)(ws + off); off += (size_t)B_ * D_ * sizeof(float);
  float* fixedc = (float*)(ws + off); off += (size_t)B_ * D_ * sizeof(float);
  int* prev = (int*)(ws + off);      off += (size_t)B_ * sizeof(int);
  unsigned char* visited = (unsigned char*)(ws + off); off += (size_t)B_ * N_;

  init_state_kernel<<<(B_ * N_ + 255) / 256, 256, 0, stream>>>(visited, prev);
  mean_kernel<<<B_, D_, 0, stream>>>(E, pref, meanE);
  // fixed_ctx = meanE @ W_fixed   (1024x256 @ 256x256), 128 cols per block
  wmma_gemm_kernel<<<dim3(B_ / 16, D_ / 128), 256, 0, stream>>>(
      meanE, Wfixed, nullptr, fixedc, D_);
  // gKVL = (embeddings + pref) @ W_node   (204800x256 @ 256x768)
  wmma_gemm_kernel<<<dim3((B_ * N_) / 16, 768 / 128), 256, 0, stream>>>(
      E, Wnode, pref, gKVL, 768);
  for (int t = 0; t < N_ - 1; ++t)
    step_kernel<<<B_ / BT, 256, 0, stream>>>(E, pref, Wstep, Wout, gKVL,
                                             fixedc, prev, visited, out, t);
}